// GiniLoss_11141145166385
// MI455X (gfx1250) — compile-verified
//
#include <hip/hip_runtime.h>
#include <stdint.h>

// ---------------------------------------------------------------------------
// GiniLoss on MI455X (gfx1250), single-workgroup, O(n log n):
//   1) TDM tensor_load_to_lds stages x (32KB) global -> LDS      [TENSORcnt]
//   2) block tree-reduction for sum (mean); min = sorted[0]
//   3) in-LDS bitonic sort (8192 = 2^13, 91 stages, 4 cmp/thread/stage)
//   4) weighted sum  S = sum_k (2k-n+1) * x_(k)  via v_wmma_f32_16x16x4_f32
//      (dot product mapped to the diagonal of the 16x16 f32 accumulator)
//   5) gini = S / (n^2 * mu),  mu = (min<0 ? mean-min : mean) + 1e-8
// ---------------------------------------------------------------------------

typedef float    v2f  __attribute__((ext_vector_type(2)));
typedef float    v8f  __attribute__((ext_vector_type(8)));
typedef uint32_t u32x4 __attribute__((ext_vector_type(4)));
typedef int      i32x8 __attribute__((ext_vector_type(8)));
typedef int      i32x4 __attribute__((ext_vector_type(4)));

#define NPOW2    8192
#define NTHREADS 1024
#define EPSV     1e-8f

__global__ __launch_bounds__(NTHREADS)
void gini_kernel(const float* __restrict__ x, float* __restrict__ out, int n) {
    __shared__ float xs[NPOW2];       // 32 KB: data, sorted in place
    __shared__ float red[NTHREADS];   //  4 KB: tree reduction
    __shared__ float dpart[8 * 32];   //  1 KB: WMMA accumulator spill
    __shared__ float s_sum;

    const int tid = threadIdx.x;

    // ---- Stage 1: TDM async tensor load x -> LDS (wave 0 issues once) ----
    if (tid < 32) {
        uint64_t ga      = (uint64_t)(uintptr_t)x;
        // low 32 bits of a flat shared-aperture pointer == LDS byte offset
        uint32_t lds_off = (uint32_t)(uintptr_t)&xs[0];

        // D# group 0 (128b): count=1 | lds_addr | global_addr[56:0] | type=2
        u32x4 g0;
        g0[0] = 1u;                                    // count=1, user mode
        g0[1] = lds_off;                               // lds_addr (bytes)
        g0[2] = (uint32_t)(ga & 0xFFFFFFFFu);          // global_addr[31:0]
        g0[3] = (uint32_t)((ga >> 32) & 0x1FFFFFFu)    // global_addr[56:32]
              | (2u << 30);                            // type = 2 ("image")

        // D# group 1 (256b): wg_mask=0, data_size=2(4B), 1-row tile of n elems
        i32x8 g1;
        g1[0] = (2 << 16);                                   // data_size = 4B
        g1[1] = (int)(((uint32_t)n & 0xFFFFu) << 16);        // tensor_dim0 lo16
        g1[2] = (int)(((uint32_t)n >> 16) | (1u << 16));     // dim0 hi, tensor_dim1=1
        g1[3] = (int)(((uint32_t)n & 0xFFFFu) << 16);        // tile_dim0 = n
        g1[4] = 1;                                           // tile_dim1 = 1
        g1[5] = n;                                           // tensor_dim0_stride
        g1[6] = 0;
        g1[7] = 0;

        i32x4 g2 = {0, 0, 0, 0};       // higher dims unused
        i32x4 g3 = {0, 0, 0, 0};
        i32x8 g4 = {0, 0, 0, 0, 0, 0, 0, 0};   // 6-arg form: extra group

        __builtin_amdgcn_tensor_load_to_lds(g0, g1, g2, g3, g4, /*cpol=*/0);
        __builtin_amdgcn_s_wait_tensorcnt(0);
    }
    __syncthreads();

    // pad (no-op when n == NPOW2): +huge sorts to the tail
    for (int i = n + tid; i < NPOW2; i += NTHREADS) xs[i] = 3.0e38f;

    // ---- Stage 2: deterministic tree reduction for sum(x) ----
    float lsum = 0.0f;
    for (int i = tid; i < n; i += NTHREADS) lsum += xs[i];
    red[tid] = lsum;
    for (int s = NTHREADS / 2; s > 0; s >>= 1) {
        __syncthreads();
        if (tid < s) red[tid] += red[tid + s];
    }
    if (tid == 0) s_sum = red[0];
    __syncthreads();

    // ---- Stage 3: bitonic sort in LDS (ascending) ----
    for (int k = 2; k <= NPOW2; k <<= 1) {
        for (int j = k >> 1; j > 0; j >>= 1) {
            __syncthreads();
            for (int i = tid; i < NPOW2; i += NTHREADS) {
                int ixj = i ^ j;
                if (ixj > i) {
                    float a = xs[i], b = xs[ixj];
                    bool up = ((i & k) == 0);
                    if (up ? (a > b) : (a < b)) { xs[i] = b; xs[ixj] = a; }
                }
            }
        }
    }
    __syncthreads();

    // ---- Stage 4: S = sum_k (2k-n+1)*x_(k) via v_wmma_f32_16x16x4_f32 ----
    // Per 64-element chunk: A[m][k] = x[4m+k] (16x4), B[k][c] = w[4c+k] (4x16),
    // so D[m][m] accumulates dot(x[4m..4m+3], w[4m..4m+3]); diagonal summed
    // across all 128 chunks gives S. 32-bit A layout (ISA 7.12.2):
    //   lanes 0-15: VGPR0=K0, VGPR1=K1 ; lanes 16-31: VGPR0=K2, VGPR1=K3.
    if (tid < 32) {
        const int lane = tid;
        const int m    = lane & 15;
        const int ofs  = (lane < 16) ? 0 : 2;
        const int lo   = 4 * m + ofs;
        const float wbase = (float)(n - 1);

        v8f acc = {};
        for (int it = 0; it < NPOW2 / 64; ++it) {
            const int g = it * 64 + lo;
            v2f a, b;
            a[0] = xs[g];
            a[1] = xs[g + 1];
            b[0] = (g     < n) ? (2.0f * (float)g        - wbase) : 0.0f;
            b[1] = (g + 1 < n) ? (2.0f * (float)(g + 1)  - wbase) : 0.0f;
            acc = __builtin_amdgcn_wmma_f32_16x16x4_f32(
                      /*neg_a=*/false, a, /*neg_b=*/false, b,
                      /*c_mod=*/(short)0, acc,
                      /*reuse_a=*/false, /*reuse_b=*/false);
        }
#pragma unroll
        for (int r = 0; r < 8; ++r) dpart[lane * 8 + r] = acc[r];
    }
    __syncthreads();

    // ---- Stage 5: extract diagonal, finalize ----
    // C/D layout: VGPR r, lanes 0-15 -> (M=r, N=lane); lanes 16-31 -> (M=r+8).
    // D[m][m]: m<8 -> lane=m, vgpr=m ; m>=8 -> lane=m+16, vgpr=m-8.
    if (tid == 0) {
        float S = 0.0f;
        for (int m = 0; m < 8; ++m)  S += dpart[m * 8 + m];
        for (int m = 8; m < 16; ++m) S += dpart[(m + 16) * 8 + (m - 8)];

        float mean = s_sum / (float)n;
        float mn   = xs[0];                       // sorted ascending
        float mu   = ((mn < 0.0f) ? (mean - mn) : mean) + EPSV;
        float nn   = (float)n * (float)n;
        out[0] = S / (nn * mu);
    }
}

extern "C" void kernel_launch(void* const* d_in, const int* in_sizes, int n_in,
                              void* d_out, int out_size, void* d_ws, size_t ws_size,
                              hipStream_t stream) {
    const float* x = (const float*)d_in[0];
    float* out     = (float*)d_out;
    int n          = in_sizes[0];
    gini_kernel<<<1, NTHREADS, 0, stream>>>(x, out, n);
}